// AllPairs_84971632984499
// MI455X (gfx1250) — compile-verified
//
#include <hip/hip_runtime.h>

// CDNA5 / gfx1250, wave32.
typedef __attribute__((ext_vector_type(2))) float v2f;
typedef __attribute__((ext_vector_type(8))) float v8f;

#define NATOMS 2048
#define NBATCH 8
#define NT 128                   // 2048 / 16 tiles per dimension
#define TILES_PER_BATCH 8256     // NT*(NT+1)/2 (upper tri incl. diagonal)
#define BLOCKS_PER_BATCH 258
#define WAVES_PER_BATCH (BLOCKS_PER_BATCH * 8)   // 2064 ; 8256/2064 = 4 tiles/wave
#define PPB 2096128              // N*(N-1)/2 pairs per batch
#define CUTOFF2 27.04f           // 5.2^2

struct F3 { float x, y, z; };    // 12B, 4B-aligned -> global_store_b96-able

// Emit one 16x16 distance tile. GUARD=true only for diagonal tiles (ti==tj),
// where slots with col <= row do not exist in the triangular output.
template <bool GUARD>
__device__ __forceinline__ void emit_tile(
    const v8f& c, const float4* crow, const float4 cj,
    int col, int rbase, int batchAtomBase, int batchPairBase,
    float* __restrict__ out_i0, float* __restrict__ out_i1,
    float* __restrict__ out_d,  F3* __restrict__ out_df)
{
    #pragma unroll
    for (int r = 0; r < 8; ++r) {
        const int row = rbase + r;                     // i0 + r (+8 for hi half)
        if (!GUARD || (col > row)) {
            const float d2 = c[r];
            float dist, dx, dy, dz, f0, f1;
            if (d2 <= CUTOFF2) {
                dist = __builtin_amdgcn_sqrtf(fmaxf(d2, 0.0f));  // raw v_sqrt_f32
                dx = crow[r].x - cj.x;
                dy = crow[r].y - cj.y;
                dz = crow[r].z - cj.z;
                f0 = (float)(batchAtomBase + row);
                f1 = (float)(batchAtomBase + col);
            } else {
                dist = 0.0f; dx = 0.0f; dy = 0.0f; dz = 0.0f;
                f0 = -1.0f;  f1 = -1.0f;
            }
            // All-int32 triangular flat index; each array has its own SGPR base,
            // so stores lower to saddr + 32-bit voffset (scale_offset) form.
            const int p = batchPairBase
                        + ((row * (2 * NATOMS - 1 - row)) >> 1)
                        + (col - row - 1);
            out_i0[p] = f0;
            out_i1[p] = f1;
            out_d[p]  = dist;
            out_df[p] = F3{dx, dy, dz};
        }
    }
}

__global__ __launch_bounds__(256) void AllPairs_wmma_kernel(
    const int* __restrict__ species,
    const float* __restrict__ coords,
    float* __restrict__ out_i0,          // indices row 0  [B*PPB]
    float* __restrict__ out_i1,          // indices row 1  [B*PPB]
    float* __restrict__ out_d,           // dist           [B*PPB]
    F3*    __restrict__ out_df)          // diff           [B*PPB, 3]
{
    // Whole batch staged in LDS: (x, y, z, r^2). 2048 * 16B = 32 KB.
    __shared__ float4 atoms[NATOMS];

    const int batch = blockIdx.y;
    const int tid   = threadIdx.x;

    const float* cb = coords + (size_t)batch * NATOMS * 3;
    const int*   sb = species + (size_t)batch * NATOMS;

    for (int a = tid; a < NATOMS; a += 256) {
        float x = cb[3 * a + 0];
        float y = cb[3 * a + 1];
        float z = cb[3 * a + 2];
        float r2 = __builtin_fmaf(x, x, __builtin_fmaf(y, y, z * z));
        // Dummy atoms (species == -1): inflate r^2 so d^2 >> cutoff^2 -> masked for free.
        if (sb[a] == -1) r2 += 1.0e30f;
        atoms[a] = make_float4(x, y, z, r2);
    }
    __syncthreads();

    const int  lane        = tid & 31;
    const int  waveInBlock = tid >> 5;
    const int  waveInBatch = blockIdx.x * 8 + waveInBlock;
    const int  m    = lane & 15;
    const bool hi   = lane >= 16;
    const int  hoff = hi ? 8 : 0;

    const int batchAtomBase = batch * NATOMS;
    const int batchPairBase = batch * PPB;

    for (int t = waveInBatch; t < TILES_PER_BATCH; t += WAVES_PER_BATCH) {
        // Invert flat tile id -> (ti, tj) with tj >= ti.
        // startT(ti) = ti*(2*NT - ti + 1)/2 ; float estimate + exact integer fixup.
        const float s = (2.0f * NT + 1.0f);
        int ti = (int)((s - __builtin_amdgcn_sqrtf(
                     __builtin_fmaf(s, s, -8.0f * (float)t))) * 0.5f);
        if (ti < 0) ti = 0;
        if (ti > NT - 1) ti = NT - 1;
        while (ti > 0 && (ti * (2 * NT - ti + 1)) / 2 > t) --ti;
        while (((ti + 1) * (2 * NT - ti)) / 2 <= t) ++ti;
        const int tj = ti + (t - (ti * (2 * NT - ti + 1)) / 2);

        const int i0 = ti * 16;
        const int j0 = tj * 16;

        // A fragment (16x4 f32): lanes 0-15 carry K0,K1 for row M=lane;
        // lanes 16-31 carry K2,K3 for row M=lane-16.  Row vector = (x, y, z, 1).
        const float4 ciA = atoms[i0 + m];
        v2f a;
        a[0] = hi ? ciA.z : ciA.x;
        a[1] = hi ? 1.0f  : ciA.y;

        // B fragment (4x16 f32), mirrored layout: col vector = (-2x, -2y, -2z, rj^2).
        const float4 cj = atoms[j0 + m];
        v2f b;
        b[0] = hi ? (-2.0f * cj.z) : (-2.0f * cj.x);
        b[1] = hi ? cj.w           : (-2.0f * cj.y);

        // C fragment preloaded with ri^2: VGPR r -> row (i0 + r) for lanes 0-15,
        // row (i0 + r + 8) for lanes 16-31.
        float4 crow[8];
        v8f c;
        #pragma unroll
        for (int r = 0; r < 8; ++r) {
            crow[r] = atoms[i0 + hoff + r];
            c[r] = crow[r].w;
        }

        // d^2 tile = ri^2 + rj^2 - 2*xi.xj  in one matrix op.
        c = __builtin_amdgcn_wmma_f32_16x16x4_f32(
                false, a, false, b, (short)0, c, false, false);

        const int col   = j0 + m;
        const int rbase = i0 + hoff;

        if (tj > ti) {
            // Off-diagonal tile: every slot is a real pair (j0 > i0+15 >= row).
            emit_tile<false>(c, crow, cj, col, rbase,
                             batchAtomBase, batchPairBase,
                             out_i0, out_i1, out_d, out_df);
        } else {
            // Diagonal tile: guard col > row.
            emit_tile<true>(c, crow, cj, col, rbase,
                            batchAtomBase, batchPairBase,
                            out_i0, out_i1, out_d, out_df);
        }
    }
}

extern "C" void kernel_launch(void* const* d_in, const int* in_sizes, int n_in,
                              void* d_out, int out_size, void* d_ws, size_t ws_size,
                              hipStream_t stream) {
    const int*   species = (const int*)d_in[0];
    const float* coords  = (const float*)d_in[1];
    float*       out     = (float*)d_out;

    const long long BP = (long long)NBATCH * PPB;
    float* out_i0 = out;                    // indices row 0
    float* out_i1 = out + BP;               // indices row 1
    float* out_d  = out + 2 * BP;           // dist
    F3*    out_df = (F3*)(out + 3 * BP);    // diff [BP,3]

    dim3 grid(BLOCKS_PER_BATCH, NBATCH);
    AllPairs_wmma_kernel<<<grid, 256, 0, stream>>>(
        species, coords, out_i0, out_i1, out_d, out_df);
}